// GraphAggregationLayer_61710090109692
// MI455X (gfx1250) — compile-verified
//
#include <hip/hip_runtime.h>
#include <hip/hip_bf16.h>

// Problem constants (match the reference exactly).
#define GB 4          // batches
#define GE 320000     // edges per batch
#define GN 10000      // nodes per batch
#define GF 64         // features
#define QPE (GF / 4)  // 16 float4-quads per edge

typedef float v4f __attribute__((ext_vector_type(4)));

// ---------------------------------------------------------------------------
// Kernel 1: zero the output accumulator (harness poisons d_out with 0xAA).
// Grid covers the output exactly (640,000 quads / 256 = 2500 blocks), so no
// bounds check is needed: one 128-bit store per thread, nothing else.
// ---------------------------------------------------------------------------
__global__ __launch_bounds__(256) void zero_out_kernel(float* __restrict__ out) {
    const int t = blockIdx.x * blockDim.x + threadIdx.x;
    v4f z = {0.0f, 0.0f, 0.0f, 0.0f};
    ((v4f*)out)[t] = z;
}

// ---------------------------------------------------------------------------
// Kernel 2: scatter-add.
//   grid = (E*QPE/256, B) = (20000, 4); exact coverage, no tail branch.
//   thread -> (edge e = t/16 within batch, quad q = t%16); a wave32 covers
//   2 consecutive edges = 512 contiguous bytes of msg_vectors -> coalesced
//   global_load_b128 with TH_NT (stream-once data must not evict the
//   L2-resident 10 MB accumulator). 4x unsafeAtomicAdd -> non-returning
//   global_atomic_add_f32 (L2 RMW units).
//   Batch comes from blockIdx.y (free SGPR), so all address math is 32-bit
//   within a batch slice.
// ---------------------------------------------------------------------------
__global__ __launch_bounds__(256) void scatter_add_kernel(
    const float* __restrict__ msg,   // [B*E*F]
    const int*   __restrict__ sidx,  // [B*E]
    float*       __restrict__ out)   // [B*N*F]
{
    const int t    = blockIdx.x * blockDim.x + threadIdx.x;  // [0, E*QPE)
    const int b    = blockIdx.y;                             // [0, B)
    const int quad = t & (QPE - 1);                          // t % 16
    const int e    = t >> 4;                                 // edge in batch

    // Per-batch base pointers (64-bit add once, then 32-bit offsets).
    const float* msg_b = msg + (long long)b * GE * GF;   // 20.48M floats/batch
    const int*   idx_b = sidx + (long long)b * GE;
    float*       out_b = out + (long long)b * GN * GF;   // 640K floats/batch

    // Node index: 16 consecutive threads hit the same dword -> L0 hit.
    const int node = idx_b[e];

    // Stream-once 128-bit load with TH_NT (coalesced: 512B per wave32).
    const v4f m = __builtin_nontemporal_load((const v4f*)msg_b + t);

    float* dst = out_b + (node * GF + quad * 4);

    // Hardware FP32 global atomics; return value unused -> non-returning form
    // (STOREcnt-tracked, no return bandwidth).
    unsafeAtomicAdd(dst + 0, m.x);
    unsafeAtomicAdd(dst + 1, m.y);
    unsafeAtomicAdd(dst + 2, m.z);
    unsafeAtomicAdd(dst + 3, m.w);
}

// ---------------------------------------------------------------------------
// Launcher
// ---------------------------------------------------------------------------
extern "C" void kernel_launch(void* const* d_in, const int* in_sizes, int n_in,
                              void* d_out, int out_size, void* d_ws, size_t ws_size,
                              hipStream_t stream) {
    const float* msg  = (const float*)d_in[0];  // [B,E,F] float32
    const int*   sidx = (const int*)  d_in[1];  // [B,E]   int32
    // d_in[2] (h_v) is unused by the reference output m_agg.
    float* out = (float*)d_out;                 // [B,N,F] float32

    // Zero the accumulator: B*N*F/4 = 640,000 quads, exactly 2500 blocks.
    {
        const int threads = 256;
        const int blocks  = (GB * GN * GF / 4) / threads;   // 2500
        zero_out_kernel<<<blocks, threads, 0, stream>>>(out);
    }

    // Scatter-add all edges: (E*QPE/256, B) = (20000, 4) blocks, exact.
    {
        const int threads = 256;
        dim3 grid(GE * QPE / threads, GB);                  // (20000, 4)
        scatter_add_kernel<<<grid, 256, 0, stream>>>(msg, sidx, out);
    }
}